// YOLOv1Loss_34737695490341
// MI455X (gfx1250) — compile-verified
//
#include <hip/hip_runtime.h>
#include <stdint.h>

// ---------------- problem constants (from reference) ----------------
#define SS            7
#define CHANS         30                         // C + B*5 = 20 + 10
#define BATCH_N       16384
#define CELLS_TOTAL   (BATCH_N * SS * SS)        // 802,816
#define TILE_CELLS    128
#define TPB           128                        // 4 waves (wave32)
#define CHUNKS        ((TILE_CELLS * CHANS * 4) / 16)   // 960 x 16B per array
#define NBLOCKS       (CELLS_TOTAL / TILE_CELLS)        // 6272, exact

// CDNA5 async global->LDS copy of one 16-byte chunk (per lane), non-temporal.
// VDST = LDS byte address (VGPR), VADDR = 64-bit global address (VGPR pair).
__device__ __forceinline__ void async_ld16(uint32_t lds_off, const void* gaddr) {
    asm volatile("global_load_async_to_lds_b128 %0, %1, off th:TH_LOAD_NT"
                 :
                 : "v"(lds_off), "v"(gaddr)
                 : "memory");
}

__device__ __forceinline__ void wait_async0() {
    asm volatile("s_wait_asynccnt 0" ::: "memory");
}

__device__ __forceinline__ float iou_yolo(float bx, float by, float bw, float bh,
                                          float cx, float cy, float cw, float ch) {
    float b1x1 = bx - bw * 0.5f, b1y1 = by - bh * 0.5f;
    float b1x2 = bx + bw * 0.5f, b1y2 = by + bh * 0.5f;
    float b2x1 = cx - cw * 0.5f, b2y1 = cy - ch * 0.5f;
    float b2x2 = cx + cw * 0.5f, b2y2 = cy + ch * 0.5f;
    float iw = fmaxf(fminf(b1x2, b2x2) - fmaxf(b1x1, b2x1), 0.0f);
    float ih = fmaxf(fminf(b1y2, b2y2) - fmaxf(b1y1, b2y1), 0.0f);
    float inter = iw * ih;
    float a1 = fabsf((b1x2 - b1x1) * (b1y2 - b1y1));
    float a2 = fabsf((b2x2 - b2x1) * (b2y2 - b2y1));
    return inter / (a1 + a2 - inter + 1e-6f);
}

__global__ __launch_bounds__(TPB)
void yolo_partial(const float* __restrict__ P, const float* __restrict__ T,
                  float* __restrict__ partial) {
    __shared__ float4 smem[2 * CHUNKS];          // [0..959]=preds tile, [960..1919]=trues tile
    const int tid = threadIdx.x;

    const size_t tileBytes = (size_t)blockIdx.x * (TILE_CELLS * CHANS * 4);
    const char* gp = (const char*)P + tileBytes;
    const char* gt = (const char*)T + tileBytes;
    const uint32_t lbase = (uint32_t)(uintptr_t)smem;

    // ---- stage tiles: coalesced async 16B/lane chunks, global -> LDS ----
    // preds chunks 0..895 : 7 uniform iterations
#pragma unroll
    for (int i = 0; i < 7; ++i) {
        int idx = i * TPB + tid;
        async_ld16(lbase + (uint32_t)idx * 16u, gp + (size_t)idx * 16);
    }
    // mixed iteration: lanes 0..63 -> preds chunks 896..959,
    //                  lanes 64..127 -> trues chunks 0..63
    {
        bool lo = tid < (CHUNKS - 7 * TPB);                  // tid < 64
        int  pidx = 7 * TPB + tid;                           // preds chunk
        int  tidx = tid - (CHUNKS - 7 * TPB);                // trues chunk
        const char* g  = lo ? (gp + (size_t)pidx * 16) : (gt + (size_t)tidx * 16);
        uint32_t    lo32 = lo ? (lbase + (uint32_t)pidx * 16u)
                              : (lbase + (uint32_t)(CHUNKS + tidx) * 16u);
        async_ld16(lo32, g);
    }
    // trues chunks 64..959 : 7 uniform iterations
#pragma unroll
    for (int i = 0; i < 7; ++i) {
        int idx = 64 + i * TPB + tid;
        async_ld16(lbase + (uint32_t)(CHUNKS + idx) * 16u, gt + (size_t)idx * 16);
    }

    wait_async0();
    __syncthreads();

    // ---- each thread computes one cell from LDS (bank-conflict-free: stride 30) ----
    const float* p = (const float*)smem + tid * CHANS;
    const float* t = (const float*)(smem + CHUNKS) + tid * CHANS;

    float t0 = t[0], t1 = t[1], t2 = t[2], t3 = t[3], t4 = t[4];
    float p0 = p[0], p1 = p[1], p2 = p[2], p3 = p[3], p4 = p[4];
    float p5 = p[5], p6 = p[6], p7 = p[7], p8 = p[8], p9 = p[9];

    float m = (t4 == 1.0f) ? 1.0f : 0.0f;

    float iou1 = iou_yolo(t0, t1, t2, t3, p0, p1, p2, p3);
    float iou2 = iou_yolo(t0, t1, t2, t3, p5, p6, p7, p8);
    bool best1 = iou1 > iou2;

    float bh0 = best1 ? p0 : p5, bh1 = best1 ? p1 : p6;
    float bh2 = best1 ? p2 : p7, bh3 = best1 ? p3 : p8;
    float Chat  = best1 ? p4 : p9;
    float Chatc = best1 ? p9 : p4;
    float Cij   = fmaxf(iou1, iou2);
    float Ccomp = fminf(iou1, iou2);

    float dx = t0 - bh0, dy = t1 - bh1;
    float xy = dx * dx + dy * dy;
    float sw = sqrtf(t2) - sqrtf(bh2);
    float sh = sqrtf(t3) - sqrtf(bh3);
    float wh = sw * sw + sh * sh;

    float dobj = Cij - Chat;
    float obj  = m * dobj * dobj + (1.0f - m) * (p4 * p4 + p9 * p9);
    float dno  = Ccomp - Chatc;

    float cls = 0.0f;
#pragma unroll
    for (int k = 10; k < CHANS; ++k) {
        float d = t[k] - p[k];
        cls += d * d;
    }

    float acc = 5.0f * m * (xy + wh) + obj + 0.5f * m * dno * dno + m * cls;

    // ---- wave32 reduction + cross-wave combine ----
#pragma unroll
    for (int off = 16; off > 0; off >>= 1)
        acc += __shfl_down(acc, off, 32);

    __shared__ float wred[TPB / 32];
    int wid = tid >> 5, lane = tid & 31;
    if (lane == 0) wred[wid] = acc;
    __syncthreads();
    if (tid == 0) {
        float s = 0.0f;
#pragma unroll
        for (int w = 0; w < TPB / 32; ++w) s += wred[w];
        partial[blockIdx.x] = s;
    }
}

__global__ __launch_bounds__(256)
void yolo_final(const float* __restrict__ partial, int n, float* __restrict__ out) {
    float acc = 0.0f;
    for (int i = threadIdx.x; i < n; i += 256) acc += partial[i];
#pragma unroll
    for (int off = 16; off > 0; off >>= 1)
        acc += __shfl_down(acc, off, 32);

    __shared__ float wred[8];
    int wid = threadIdx.x >> 5, lane = threadIdx.x & 31;
    if (lane == 0) wred[wid] = acc;
    __syncthreads();
    if (threadIdx.x == 0) {
        float s = 0.0f;
#pragma unroll
        for (int w = 0; w < 8; ++w) s += wred[w];
        out[0] = s * (1.0f / (float)BATCH_N);
    }
}

extern "C" void kernel_launch(void* const* d_in, const int* in_sizes, int n_in,
                              void* d_out, int out_size, void* d_ws, size_t ws_size,
                              hipStream_t stream) {
    const float* P = (const float*)d_in[0];   // y_preds
    const float* T = (const float*)d_in[1];   // y_trues
    float* partial = (float*)d_ws;            // 6272 floats of scratch

    yolo_partial<<<NBLOCKS, TPB, 0, stream>>>(P, T, partial);
    yolo_final<<<1, 256, 0, stream>>>(partial, NBLOCKS, (float*)d_out);
}